// DilateLossCUDA_11536282157288
// MI455X (gfx1250) — compile-verified
//
#include <hip/hip_runtime.h>

#define BB   256
#define NN   336
#define NP   337          // N+1 (R matrix with border)
#define VV   7
#define BIGF 1e8f
#define NTILE 21          // 336/16

typedef float v2f __attribute__((ext_vector_type(2)));
typedef float v8f __attribute__((ext_vector_type(8)));

// ---------------------------------------------------------------------------
// Kernel 1: D[b,i,j] = max(|T_i|^2 + |O_j|^2 - 2*T_i.O_j, 0)
// One wave32 per 16x16 tile; cross term via V_WMMA_F32_16X16X4_F32 (K=7 -> 2x K4)
// ---------------------------------------------------------------------------
__global__ __launch_bounds__(256) void pairdist_wmma(
    const float* __restrict__ T, const float* __restrict__ O,
    float* __restrict__ D) {
  int wave = (blockIdx.x * blockDim.x + threadIdx.x) >> 5;
  int lane = threadIdx.x & 31;

  int b = wave / (NTILE * NTILE);
  if (b >= BB) return;                       // wave-uniform, EXEC stays full
  int rem = wave % (NTILE * NTILE);
  int ti = rem / NTILE, tj = rem % NTILE;

  const float* Tb = T + (size_t)b * NN * VV;
  const float* Ob = O + (size_t)b * NN * VV;

  int m  = ti * 16 + (lane & 15);            // A row (targets)
  int n  = tj * 16 + (lane & 15);            // B col (outputs)
  int kh = (lane < 16) ? 0 : 2;              // per-ISA f32 A/B K split

  v2f a0, b0, a1, b1;
  a0.x = Tb[m * VV + kh];     a0.y = Tb[m * VV + kh + 1];
  b0.x = Ob[n * VV + kh];     b0.y = Ob[n * VV + kh + 1];
  int k2 = 4 + kh;                           // 4 or 6; k2+1==7 is the pad lane
  a1.x = Tb[m * VV + k2];
  b1.x = Ob[n * VV + k2];
  float ay = 0.0f, by = 0.0f;
  if (k2 + 1 < VV) { ay = Tb[m * VV + k2 + 1]; by = Ob[n * VV + k2 + 1]; }
  a1.y = ay;  b1.y = by;

  v8f c = {};
  c = __builtin_amdgcn_wmma_f32_16x16x4_f32(false, a0, false, b0, (short)0, c,
                                            false, false);
  c = __builtin_amdgcn_wmma_f32_16x16x4_f32(false, a1, false, b1, (short)0, c,
                                            false, false);

  // norms
  float on = 0.0f;
#pragma unroll
  for (int v = 0; v < VV; ++v) { float t = Ob[n * VV + v]; on += t * t; }

  int rowBase = ti * 16 + ((lane < 16) ? 0 : 8);
  int col     = tj * 16 + (lane & 15);
  float* Db = D + (size_t)b * NN * NN;
#pragma unroll
  for (int r = 0; r < 8; ++r) {
    int row = rowBase + r;
    float tn = 0.0f;
#pragma unroll
    for (int v = 0; v < VV; ++v) { float t = Tb[row * VV + v]; tn += t * t; }
    float d = tn + on - 2.0f * c[r];
    Db[(size_t)row * NN + col] = fmaxf(d, 0.0f);
  }
}

// ---------------------------------------------------------------------------
// Kernel 2: soft-DTW forward. One block per batch, anti-diagonal wavefront.
// R[i,j] = D[i-1,j-1] + softmin(R[i-1,j-1], R[i-1,j], R[i,j-1]); gamma = 1.
// ---------------------------------------------------------------------------
__global__ __launch_bounds__(352) void dtw_forward(
    const float* __restrict__ D, float* __restrict__ R) {
  __shared__ float buf[3][NP];
  int b = blockIdx.x;
  int i = threadIdx.x;
  const float* Db = D + (size_t)b * NN * NN;
  float* Rb = R + (size_t)b * NP * NP;

  for (int d = 0; d <= 2 * NN; ++d) {
    float* cur       = buf[d % 3];
    const float* p1  = buf[(d + 2) % 3];   // diagonal d-1
    const float* p2  = buf[(d + 1) % 3];   // diagonal d-2
    int j = d - i;
    if (i < NP && j >= 0 && j < NP) {
      float v;
      if (i == 0)       v = (j == 0) ? 0.0f : BIGF;
      else if (j == 0)  v = BIGF;
      else {
        float rd = p2[i - 1], ru = p1[i - 1], rl = p1[i];
        float mn = fminf(rd, fminf(ru, rl));
        float z  = __expf(mn - rd) + __expf(mn - ru) + __expf(mn - rl);
        v = Db[(size_t)(i - 1) * NN + (j - 1)] + (mn - __logf(z));
        if (j + 1 < NN)  // warm L2/L1 for the diagonal two steps ahead
          __builtin_prefetch(&Db[(size_t)(i - 1) * NN + (j + 1)], 0, 1);
      }
      cur[i] = v;
      Rb[(size_t)i * NP + j] = v;
    }
    __syncthreads();
  }
}

// ---------------------------------------------------------------------------
// Kernel 3: soft-DTW backward (gradient E) fused with  sum(E * (i-j)^2).
// Reverse wavefront; per-batch partial written to accT[b] (deterministic).
// ---------------------------------------------------------------------------
__global__ __launch_bounds__(352) void dtw_backward(
    const float* __restrict__ D, const float* __restrict__ R,
    float* __restrict__ accT) {
  __shared__ float buf[3][NP];
  __shared__ float red[352];
  int b = blockIdx.x;
  int t = threadIdx.x;
  const float* Db = D + (size_t)b * NN * NN;
  const float* Rb = R + (size_t)b * NP * NP;

  for (int k = t; k < 3 * NP; k += blockDim.x) ((float*)buf)[k] = 0.0f;
  __syncthreads();

  float lacc = 0.0f;
  int i = t;
  for (int s = 2 * NN; s >= 2; --s) {
    float* cur       = buf[s % 3];
    const float* e1  = buf[(s + 1) % 3];   // diagonal s+1
    const float* e2  = buf[(s + 2) % 3];   // diagonal s+2
    int j = s - i;
    if (i < NP) {
      float E = 0.0f;
      if (i >= 1 && j >= 1 && j <= NN) {
        float Rij = Rb[(size_t)i * NP + j];
        float aw = 0.0f, bw = 0.0f, cw = 0.0f;
        if (i < NN)
          aw = __expf(Rb[(size_t)(i + 1) * NP + j] - Rij -
                      Db[(size_t)i * NN + (j - 1)]) * e1[i + 1];
        if (j < NN)
          bw = __expf(Rb[(size_t)i * NP + (j + 1)] - Rij -
                      Db[(size_t)(i - 1) * NN + j]) * e1[i];
        if (i < NN && j < NN)
          cw = __expf(Rb[(size_t)(i + 1) * NP + (j + 1)] - Rij -
                      Db[(size_t)i * NN + j]) * e2[i + 1];
        else if (i == NN && j == NN)
          cw = 1.0f;                       // E[N,N] seed
        E = aw + bw + cw;
        float dd = (float)(i - j);
        lacc += E * dd * dd;
      }
      cur[i] = E;
    }
    __syncthreads();
  }

  red[t] = lacc;
  __syncthreads();
  for (int off = 176; off >= 11; off >>= 1) {   // 352 = 2^5 * 11
    if (t < off) red[t] += red[t + off];
    __syncthreads();
  }
  if (t == 0) {
    float s = 0.0f;
    for (int k = 0; k < 11; ++k) s += red[k];
    accT[b] = s;
  }
}

// ---------------------------------------------------------------------------
// Kernel 4: deterministic final combine.
// loss = 0.5 * mean_b R[N,N]  +  0.5 * (sum E_std*omega) / (N*N*B*B)
// (E_jax = E_std/B because _soft_dtw returns sum/B.)
// ---------------------------------------------------------------------------
__global__ void finalize(const float* __restrict__ R,
                         const float* __restrict__ accT,
                         float* __restrict__ out) {
  float sS = 0.0f, sT = 0.0f;
  for (int b = 0; b < BB; ++b) {
    sS += R[(size_t)b * NP * NP + (size_t)NN * NP + NN];
    sT += accT[b];
  }
  float loss_shape = sS / (float)BB;
  float denom = (float)NN * (float)NN * (float)BB * (float)BB;
  float loss_temporal = sT / denom;
  out[0] = 0.5f * loss_shape + 0.5f * loss_temporal;
}

// ---------------------------------------------------------------------------
extern "C" void kernel_launch(void* const* d_in, const int* in_sizes, int n_in,
                              void* d_out, int out_size, void* d_ws,
                              size_t ws_size, hipStream_t stream) {
  const float* outputs = (const float*)d_in[0];  // (B,N,V)
  const float* targets = (const float*)d_in[1];  // (B,N,V)

  char* ws = (char*)d_ws;
  size_t D_bytes = (size_t)BB * NN * NN * sizeof(float);   // ~115.6 MB
  size_t R_bytes = (size_t)BB * NP * NP * sizeof(float);   // ~116.3 MB
  float* Dw   = (float*)ws;
  float* Rw   = (float*)(ws + D_bytes);
  float* accT = (float*)(ws + D_bytes + R_bytes);          // BB floats

  // D tiles: 21x21 per batch, one wave per tile, 8 waves per block
  int nWaves  = BB * NTILE * NTILE;          // 112896
  int nBlocks = nWaves / 8;                  // exact
  hipLaunchKernelGGL(pairdist_wmma, dim3(nBlocks), dim3(256), 0, stream,
                     targets, outputs, Dw);
  hipLaunchKernelGGL(dtw_forward, dim3(BB), dim3(352), 0, stream, Dw, Rw);
  hipLaunchKernelGGL(dtw_backward, dim3(BB), dim3(352), 0, stream, Dw, Rw, accT);
  hipLaunchKernelGGL(finalize, dim3(1), dim3(1), 0, stream, Rw, accT,
                     (float*)d_out);
}